// RM_SEW_only_CA_37503654428916
// MI455X (gfx1250) — compile-verified
//
#include <hip/hip_runtime.h>
#include <hip/hip_bf16.h>
#include <math.h>

// Problem constants (from reference): x (16,4,128,64,64) f32
#define NB   16
#define NF   4
#define NC   128
#define HW   4096      // 64*64
#define FHW  16384     // 4*64*64
#define TOPK 103       // ceil(0.8 * 128)

typedef __attribute__((ext_vector_type(2))) float v2f;
typedef __attribute__((ext_vector_type(4))) float v4f;
typedef __attribute__((ext_vector_type(8))) float v8f;

// ---------------------------------------------------------------------------
// Kernel A: per-(b,c) sum & max over (f,h,w). One block per (b,c): 2048 blocks.
// Reads 64KB contiguous-ish (4 chunks of 16KB) per block with B128 loads.
// ---------------------------------------------------------------------------
__global__ void __launch_bounds__(256)
ca_reduce_kernel(const float* __restrict__ x,
                 float* __restrict__ avg_out,
                 float* __restrict__ max_out) {
  __shared__ float s_sum[256];
  __shared__ float s_max[256];
  const int bc  = blockIdx.x;      // 0..2047
  const int b   = bc >> 7;
  const int c   = bc & 127;
  const int tid = threadIdx.x;

  float sum = 0.0f;
  float mx  = -INFINITY;
  for (int f = 0; f < NF; ++f) {
    const v4f* p = (const v4f*)(x + ((size_t)((b * NF + f) * NC + c) * HW));
    for (int j = tid; j < HW / 4; j += 256) {
      v4f v = p[j];
      sum += v.x + v.y + v.z + v.w;
      mx = fmaxf(mx, fmaxf(fmaxf(v.x, v.y), fmaxf(v.z, v.w)));
    }
  }
  s_sum[tid] = sum;
  s_max[tid] = mx;
  __syncthreads();
  for (int off = 128; off > 0; off >>= 1) {
    if (tid < off) {
      s_sum[tid] += s_sum[tid + off];
      s_max[tid]  = fmaxf(s_max[tid], s_max[tid + off]);
    }
    __syncthreads();
  }
  if (tid == 0) {
    avg_out[bc] = s_sum[0] * (1.0f / (float)FHW);
    max_out[bc] = s_max[0];
  }
}

// ---------------------------------------------------------------------------
// Kernel B: MLP via V_WMMA_F32_16X16X4_F32 + sigmoid + top-k -> scale[b,c].
// One block, 256 threads = 8 waves; wave w owns N-tile [16w, 16w+16).
//
// fp32 WMMA layouts (ISA 7.12.2, wave32):
//   A (16x4):  lane L: m = L%16, half = L/16; a.x = A[m][k0+2*half], a.y = +1
//   B (4x16):  lane L: n = L%16;              b.x = B[k0+2*half][n], b.y = +1
//              (B[k][n] = W[n][k] for v @ W^T, so b = two consecutive elems
//               of row n of W -- contiguous loads)
//   C/D (16x16): lane L: n = L%16, VGPR v holds m = v + 8*(L/16)
// ---------------------------------------------------------------------------
__global__ void __launch_bounds__(256)
ca_mlp_topk_kernel(const float* __restrict__ w1,
                   const float* __restrict__ w2,
                   const float* __restrict__ avg_in,
                   const float* __restrict__ max_in,
                   float* __restrict__ scale_out) {
  __shared__ float h_avg[16 * 128];
  __shared__ float h_mx [16 * 128];
  __shared__ float ca_s [16 * 128];

  const int tid    = threadIdx.x;
  const int lane   = tid & 31;
  const int wave   = tid >> 5;        // 0..7  -> N tile index
  const int half   = lane >> 4;       // 0/1
  const int mn     = lane & 15;       // m (for A) / n (for B,C,D)
  const int n_glob = wave * 16 + mn;  // global output column for this lane

  // ---- layer 1: h = relu(v @ w1^T), v in {avg, max} ----
  v8f acc_a = {0.f, 0.f, 0.f, 0.f, 0.f, 0.f, 0.f, 0.f};
  v8f acc_m = {0.f, 0.f, 0.f, 0.f, 0.f, 0.f, 0.f, 0.f};
  for (int k0 = 0; k0 < 128; k0 += 4) {
    const int ka = k0 + 2 * half;
    v2f a1, a2, bb;
    a1.x = avg_in[mn * 128 + ka];     a1.y = avg_in[mn * 128 + ka + 1];
    a2.x = max_in[mn * 128 + ka];     a2.y = max_in[mn * 128 + ka + 1];
    bb.x = w1[n_glob * 128 + ka];     bb.y = w1[n_glob * 128 + ka + 1];
    acc_a = __builtin_amdgcn_wmma_f32_16x16x4_f32(false, a1, false, bb,
                                                  (short)0, acc_a, false, false);
    acc_m = __builtin_amdgcn_wmma_f32_16x16x4_f32(false, a2, false, bb,
                                                  (short)0, acc_m, false, false);
  }
#pragma unroll
  for (int v = 0; v < 8; ++v) {
    const int m = v + 8 * half;
    h_avg[m * 128 + n_glob] = fmaxf(acc_a[v], 0.0f);
    h_mx [m * 128 + n_glob] = fmaxf(acc_m[v], 0.0f);
  }
  __syncthreads();

  // ---- layer 2: y = h_avg @ w2^T + h_mx @ w2^T ; ca = sigmoid(y) ----
  v8f acc = {0.f, 0.f, 0.f, 0.f, 0.f, 0.f, 0.f, 0.f};
  for (int k0 = 0; k0 < 128; k0 += 4) {
    const int ka = k0 + 2 * half;
    v2f a1, a2, bb;
    a1.x = h_avg[mn * 128 + ka];      a1.y = h_avg[mn * 128 + ka + 1];
    a2.x = h_mx [mn * 128 + ka];      a2.y = h_mx [mn * 128 + ka + 1];
    bb.x = w2[n_glob * 128 + ka];     bb.y = w2[n_glob * 128 + ka + 1];
    acc = __builtin_amdgcn_wmma_f32_16x16x4_f32(false, a1, false, bb,
                                                (short)0, acc, false, false);
    acc = __builtin_amdgcn_wmma_f32_16x16x4_f32(false, a2, false, bb,
                                                (short)0, acc, false, false);
  }
#pragma unroll
  for (int v = 0; v < 8; ++v) {
    const int m = v + 8 * half;
    const float y = acc[v];
    ca_s[m * 128 + n_glob] = 1.0f / (1.0f + expf(-y));
  }
  __syncthreads();

  // ---- top-103 per row (rank with index tie-break, matches lax.top_k) ----
  for (int e = tid; e < NB * NC; e += 256) {
    const int b = e >> 7;
    const int c = e & 127;
    const float v = ca_s[b * 128 + c];
    int cnt = 0;
    for (int j = 0; j < 128; ++j) {
      const float o = ca_s[b * 128 + j];
      cnt += (o > v) || (o == v && j < c);
    }
    scale_out[e] = (cnt < TOPK) ? v : 0.0f;   // mask * wta == ca * mask
  }
}

// ---------------------------------------------------------------------------
// Kernel C: out = x * scale[b,c]. One block per 4096-float (b,f,c) chunk
// (8192 blocks). x read (expected L2-resident), out written non-temporal so
// the output stream doesn't evict x from the 192MB L2.
// ---------------------------------------------------------------------------
__global__ void __launch_bounds__(256)
ca_scale_kernel(const float* __restrict__ x,
                const float* __restrict__ scale,
                float* __restrict__ out) {
  const int chunk = blockIdx.x;          // (b*4 + f)*128 + c
  const int c = chunk & 127;
  const int b = chunk >> 9;              // / (4*128)
  const float s = scale[b * NC + c];
  const size_t base = (size_t)chunk * HW;
  const v4f* px = (const v4f*)(x + base);
  v4f*       po = (v4f*)(out + base);
  for (int j = threadIdx.x; j < HW / 4; j += 256) {
    v4f v = px[j];
    v4f r = v * s;
    __builtin_nontemporal_store(r, po + j);
  }
}

extern "C" void kernel_launch(void* const* d_in, const int* in_sizes, int n_in,
                              void* d_out, int out_size, void* d_ws, size_t ws_size,
                              hipStream_t stream) {
  const float* x  = (const float*)d_in[0];
  const float* w1 = (const float*)d_in[1];
  const float* w2 = (const float*)d_in[2];
  float* out = (float*)d_out;

  float* ws    = (float*)d_ws;
  float* avg   = ws;                 // 2048 floats
  float* mx    = ws + NB * NC;       // 2048 floats
  float* scale = ws + 2 * NB * NC;   // 2048 floats

  ca_reduce_kernel  <<<NB * NC,      256, 0, stream>>>(x, avg, mx);
  ca_mlp_topk_kernel<<<1,            256, 0, stream>>>(w1, w2, avg, mx, scale);
  ca_scale_kernel   <<<NB * NF * NC, 256, 0, stream>>>(x, scale, out);
}